// HashEncoder_17514876634258
// MI455X (gfx1250) — compile-verified
//
#include <hip/hip_runtime.h>
#include <stdint.h>

#define NLEV 16
#define TBL_N 4096            // only first 4096 entries of every table are ever indexed (HASH_MOD quirk)
#define HASH_MASK 4095u
#define BLOCK 256
#define PPT 8                 // points per thread -> 2048 points per block
#define PRIME_Y 2654435761u
#define PRIME_Z 805459861u

struct TablePtrs { const float* t[NLEV]; };

// ---- CDNA5 async global->LDS (ASYNCcnt path) ----------------------------
typedef int v4i __attribute__((vector_size(16)));
typedef __attribute__((address_space(1))) v4i* g_v4i_ptr;  // global
typedef __attribute__((address_space(3))) v4i* l_v4i_ptr;  // LDS

__device__ __forceinline__ void async_lds_b128(const void* g, void* l) {
#if __has_builtin(__builtin_amdgcn_global_load_async_to_lds_b128)
  __builtin_amdgcn_global_load_async_to_lds_b128(
      (g_v4i_ptr)(uintptr_t)g,
      (l_v4i_ptr)(uint32_t)(uintptr_t)l,   // low 32 bits of generic LDS ptr == LDS offset
      0, 0);
#else
  uint32_t loff = (uint32_t)(uintptr_t)l;
  uint64_t ga   = (uint64_t)(uintptr_t)g;
  asm volatile("global_load_async_to_lds_b128 %0, %1, off"
               :: "v"(loff), "v"(ga) : "memory");
#endif
}

__device__ __forceinline__ void wait_async0() {
#if __has_builtin(__builtin_amdgcn_s_wait_asynccnt)
  __builtin_amdgcn_s_wait_asynccnt(0);
#else
  asm volatile("s_wait_asynccnt 0" ::: "memory");
#endif
}

// copy 32KB (4096 x float2) table slice into LDS: 2048 b128 chunks, 8 per thread
__device__ __forceinline__ void copy_table_async(const float* __restrict__ gsrc,
                                                 float2* ldst, int tid) {
#pragma unroll
  for (int i = 0; i < 8; ++i) {
    int c = tid + i * BLOCK;  // 16-byte chunk id; consecutive lanes -> consecutive 16B
    async_lds_b128((const char*)gsrc + c * 16, (char*)ldst + c * 16);
  }
}

__device__ __forceinline__ int iclamp(int v, int lo, int hi) {
  return v < lo ? lo : (v > hi ? hi : v);
}

__launch_bounds__(BLOCK, 1)
__global__ void hashgrid_encode_kernel(const float* __restrict__ pos,
                                       TablePtrs tp,
                                       float* __restrict__ out,
                                       int npoints) {
  __shared__ float2 tab[2][TBL_N];   // 2 x 32KB double buffer
  const int tid  = threadIdx.x;
  const int base = (int)blockIdx.x * (BLOCK * PPT);   // p, p*3, p*32 all fit in int32

  // Load positions once; hoist level-invariant (p+1)*0.5 (matches reference association:
  // ((p + 1) * 0.5) * (res - 1) ).
  float tx[PPT], ty[PPT], tz[PPT];
#pragma unroll
  for (int k = 0; k < PPT; ++k) {
    int p = base + tid + k * BLOCK;
    uint32_t b = (p < npoints) ? (uint32_t)p * 3u : 0u;
    tx[k] = (pos[b + 0] + 1.0f) * 0.5f;
    ty[k] = (pos[b + 1] + 1.0f) * 0.5f;
    tz[k] = (pos[b + 2] + 1.0f) * 0.5f;
  }

  // Prologue: stage level 0 table
  copy_table_async(tp.t[0], &tab[0][0], tid);
  wait_async0();
  __syncthreads();

#pragma unroll 1
  for (int L = 0; L < NLEV; ++L) {
    const int cur = L & 1;
    // Prefetch next level's table into the other buffer (overlapped with compute)
    if (L + 1 < NLEV) copy_table_async(tp.t[L + 1], &tab[1 - cur][0], tid);

    const int   resm1  = (16 << L) - 1;        // ceil(16*2^L) - 1, exact for SCALE=2
    const float resm1f = (float)resm1;

#pragma unroll 2
    for (int k = 0; k < PPT; ++k) {
      const int p = base + tid + k * BLOCK;

      float sx = tx[k] * resm1f;
      float sy = ty[k] * resm1f;
      float sz = tz[k] * resm1f;
      float fx = floorf(sx), fy = floorf(sy), fz = floorf(sz);
      float wx = sx - fx,    wy = sy - fy,    wz = sz - fz;
      int gx = (int)fx, gy = (int)fy, gz = (int)fz;

      int x0 = iclamp(gx,     0, resm1), x1 = iclamp(gx + 1, 0, resm1);
      int y0 = iclamp(gy,     0, resm1), y1 = iclamp(gy + 1, 0, resm1);
      int z0 = iclamp(gz,     0, resm1), z1 = iclamp(gz + 1, 0, resm1);

      // hash: x*1 ^ y*P1 ^ z*P2, then % 4096 == & 4095
      uint32_t hx0 = (uint32_t)x0,           hx1 = (uint32_t)x1;
      uint32_t hy0 = (uint32_t)y0 * PRIME_Y, hy1 = (uint32_t)y1 * PRIME_Y;
      uint32_t hz0 = (uint32_t)z0 * PRIME_Z, hz1 = (uint32_t)z1 * PRIME_Z;

      float2 f000 = tab[cur][(hx0 ^ hy0 ^ hz0) & HASH_MASK];
      float2 f001 = tab[cur][(hx0 ^ hy0 ^ hz1) & HASH_MASK];
      float2 f010 = tab[cur][(hx0 ^ hy1 ^ hz0) & HASH_MASK];
      float2 f011 = tab[cur][(hx0 ^ hy1 ^ hz1) & HASH_MASK];
      float2 f100 = tab[cur][(hx1 ^ hy0 ^ hz0) & HASH_MASK];
      float2 f101 = tab[cur][(hx1 ^ hy0 ^ hz1) & HASH_MASK];
      float2 f110 = tab[cur][(hx1 ^ hy1 ^ hz0) & HASH_MASK];
      float2 f111 = tab[cur][(hx1 ^ hy1 ^ hz1) & HASH_MASK];

      // Reference quirk preserved: dz axis lerped with wx, dx axis with wz.
      float omx = 1.0f - wx, omy = 1.0f - wy, omz = 1.0f - wz;
      float c00x = f000.x * omx + f001.x * wx, c00y = f000.y * omx + f001.y * wx;
      float c01x = f010.x * omx + f011.x * wx, c01y = f010.y * omx + f011.y * wx;
      float c10x = f100.x * omx + f101.x * wx, c10y = f100.y * omx + f101.y * wx;
      float c11x = f110.x * omx + f111.x * wx, c11y = f110.y * omx + f111.y * wx;
      float c0x = c00x * omy + c01x * wy, c0y = c00y * omy + c01y * wy;
      float c1x = c10x * omy + c11x * wy, c1y = c10y * omy + c11y * wy;
      float2 r;
      r.x = c0x * omz + c1x * wz;
      r.y = c0y * omz + c1y * wz;

      if (p < npoints) {
        uint32_t o = (uint32_t)p * (2u * NLEV) + 2u * (uint32_t)L;
        *reinterpret_cast<float2*>(out + o) = r;   // global_store_b64, GVS addressing
      }
    }

    if (L + 1 < NLEV) {
      wait_async0();      // next level's table fully in LDS
      __syncthreads();    // also guards buffer reuse next iteration
    }
  }
}

extern "C" void kernel_launch(void* const* d_in, const int* in_sizes, int n_in,
                              void* d_out, int out_size, void* d_ws, size_t ws_size,
                              hipStream_t stream) {
  const float* pos = (const float*)d_in[0];
  TablePtrs tp;
  for (int i = 0; i < NLEV; ++i) tp.t[i] = (const float*)d_in[1 + i];
  float* out = (float*)d_out;

  const int npoints = in_sizes[0] / 3;                      // 2,097,152
  const int per_block = BLOCK * PPT;                        // 2048
  const int blocks = (npoints + per_block - 1) / per_block; // 1024

  hipLaunchKernelGGL(hashgrid_encode_kernel, dim3(blocks), dim3(BLOCK), 0, stream,
                     pos, tp, out, npoints);
}